// RoGPELinearNodeEncoder_80281528697034
// MI455X (gfx1250) — compile-verified
//
#include <hip/hip_runtime.h>

// ---------------------------------------------------------------------------
// Problem constants
#define NNODES 4096
#define NEDGES 131072
#define INDIM  128
#define NW     128            // bitset words per row (4096 bits)
#define LDP    132            // padded LDS row stride (floats) to avoid bank conflicts
// ---------------------------------------------------------------------------

typedef float v2f __attribute__((ext_vector_type(2)));
typedef float v8f __attribute__((ext_vector_type(8)));

// ---------------------------------------------------------------------------
// Fused 3-layer MLP (Linear+ReLU x3) + head dot-product, per 128-row tile.
// Uses V_WMMA_F32_16X16X4_F32 (full fp32 matrix pipe, no downconvert).
// Block: 256 threads = 8 waves; wave w owns rows [16w,16w+16) of the tile.
// ---------------------------------------------------------------------------
__global__ __launch_bounds__(256)
void mlp_kernel(const float* __restrict__ coeffs,
                const float* __restrict__ W0, const float* __restrict__ b0,
                const float* __restrict__ W1, const float* __restrict__ W2,
                const float* __restrict__ W3, const float* __restrict__ b3,
                float* __restrict__ Xraw)
{
    __shared__ float sh_in[INDIM * LDP];  // activation tile [128][132]
    __shared__ float sh_w [INDIM * LDP];  // W^T staged     [k][o]

    const int tid  = threadIdx.x;
    const int wave = tid >> 5;
    const int lane = tid & 31;
    const int rowBase = blockIdx.x * 128;

    // Load input tile (coalesced rows)
    for (int i = tid; i < INDIM * INDIM; i += 256) {
        int r = i >> 7, c = i & 127;
        sh_in[r * LDP + c] = coeffs[(size_t)(rowBase + r) * INDIM + c];
    }

    const float* Ws[3] = { W0, W1, W2 };

    for (int layer = 0; layer < 3; ++layer) {
        __syncthreads();                      // sh_in writes done, old sh_w reads done
        const float* W = Ws[layer];
        for (int i = tid; i < INDIM * INDIM; i += 256) {
            int o = i >> 7, k = i & 127;
            sh_w[k * LDP + o] = W[i];         // transpose-stage: Wt[k][o] = W[o][k]
        }
        __syncthreads();

        // Accumulators: 8 col-tiles of 16x16 f32 (C/D: lane -> N, vgpr -> M)
        v8f acc[8];
        const int ncol = lane & 15;
        #pragma unroll
        for (int ct = 0; ct < 8; ++ct) {
            float bv = (layer == 0) ? b0[ct * 16 + ncol] : 0.0f;
            acc[ct] = (v8f){bv, bv, bv, bv, bv, bv, bv, bv};
        }

        const int rA   = wave * 16 + ncol;    // A frag: lane -> M
        const int kOff = (lane >> 4) << 1;    // lanes 0-15: K+0/K+1, lanes 16-31: K+2/K+3

        for (int kt = 0; kt < 32; ++kt) {     // K = 128 in chunks of 4
            const int kk = kt * 4 + kOff;
            v2f a;
            a.x = sh_in[rA * LDP + kk];
            a.y = sh_in[rA * LDP + kk + 1];
            #pragma unroll
            for (int ct = 0; ct < 8; ++ct) {
                v2f b;
                b.x = sh_w[(kk)     * LDP + ct * 16 + ncol];
                b.y = sh_w[(kk + 1) * LDP + ct * 16 + ncol];
                acc[ct] = __builtin_amdgcn_wmma_f32_16x16x4_f32(
                    false, a, false, b, (short)0, acc[ct], false, false);
            }
        }

        // ReLU + write back into this wave's own rows (no cross-wave hazard)
        #pragma unroll
        for (int ct = 0; ct < 8; ++ct) {
            #pragma unroll
            for (int v = 0; v < 8; ++v) {
                int r = wave * 16 + v + ((lane >> 4) << 3);
                int c = ct * 16 + ncol;
                sh_in[r * LDP + c] = fmaxf(acc[ct][v], 0.0f);
            }
        }
    }
    __syncthreads();

    // Head: X = h @ W3^T + b3  (OUT = 1)
    if (tid < 128) {
        float s = b3[0];
        #pragma unroll 4
        for (int k = 0; k < INDIM; ++k)
            s += sh_in[tid * LDP + k] * W3[k];
        Xraw[rowBase + tid] = s;
    }
}

// ---------------------------------------------------------------------------
// BatchNorm1d over the batch dim (single feature): training-mode batch stats.
// One block so stats + normalize fuse with a single __syncthreads.
// ---------------------------------------------------------------------------
__global__ __launch_bounds__(1024)
void bn_kernel(const float* __restrict__ Xraw, float* __restrict__ X,
               const float* __restrict__ gamma, const float* __restrict__ beta)
{
    __shared__ float s_sum[32], s_sq[32];
    const int tid = threadIdx.x;
    float s = 0.f, q = 0.f;
    for (int i = tid; i < NNODES; i += 1024) {
        float v = Xraw[i]; s += v; q += v * v;
    }
    for (int o = 16; o; o >>= 1) { s += __shfl_down(s, o); q += __shfl_down(q, o); }
    if ((tid & 31) == 0) { s_sum[tid >> 5] = s; s_sq[tid >> 5] = q; }
    __syncthreads();
    if (tid < 32) {
        s = s_sum[tid]; q = s_sq[tid];
        for (int o = 16; o; o >>= 1) { s += __shfl_down(s, o); q += __shfl_down(q, o); }
        if (tid == 0) { s_sum[0] = s; s_sq[0] = q; }
    }
    __syncthreads();
    const float m   = s_sum[0] * (1.0f / NNODES);
    const float var = s_sq[0] * (1.0f / NNODES) - m * m;
    const float inv = rsqrtf(var + 1e-5f);
    const float g = gamma[0], b = beta[0];
    for (int i = tid; i < NNODES; i += 1024)
        X[i] = (Xraw[i] - m) * inv * g + b;
}

// ---------------------------------------------------------------------------
// Build boolean adjacency bitset: A[row] bit col  (dedup is free: OR idempotent)
// ---------------------------------------------------------------------------
__global__ __launch_bounds__(256)
void build_adj(const int* __restrict__ ei, unsigned* __restrict__ A)
{
    int e = blockIdx.x * 256 + threadIdx.x;
    if (e >= NEDGES) return;
    int r = ei[e];
    int c = ei[NEDGES + e];
    atomicOr(&A[(size_t)r * NW + (c >> 5)], 1u << (c & 31));
}

// ---------------------------------------------------------------------------
// Boolean compose: dst_row[i] = OR_{j set in mask_row[i]} src_row[j].
//   A2 = compose(mask=A, src=A);  A3 = compose(mask=A, src=A2)  (== bool(A^3))
// One wave per row; mask words broadcast wave-uniform via shfl; each lane ORs
// a coalesced uint4 (16B) slice of the source row.
// ---------------------------------------------------------------------------
__global__ __launch_bounds__(256)
void bool_compose(const unsigned* __restrict__ mask,
                  const unsigned* __restrict__ src,
                  unsigned* __restrict__ dst)
{
    const int wave = threadIdx.x >> 5;
    const int lane = threadIdx.x & 31;
    const int row  = blockIdx.x * 8 + wave;

    const uint4 mw = ((const uint4*)(mask + (size_t)row * NW))[lane];
    unsigned mwv[4] = { mw.x, mw.y, mw.z, mw.w };
    uint4 acc = { 0u, 0u, 0u, 0u };

    #pragma unroll
    for (int q = 0; q < 4; ++q) {
        for (int ls = 0; ls < 32; ++ls) {
            unsigned word = (unsigned)__shfl((int)mwv[q], ls);  // wave-uniform
            while (word) {
                int b = __builtin_ctz(word);
                word &= word - 1;
                int j = ((ls * 4 + q) << 5) + b;
                uint4 v = ((const uint4*)(src + (size_t)j * NW))[lane];
                acc.x |= v.x; acc.y |= v.y; acc.z |= v.z; acc.w |= v.w;
            }
        }
    }
    ((uint4*)(dst + (size_t)row * NW))[lane] = acc;
}

// ---------------------------------------------------------------------------
// Bitset matrix-vector hop with fused step/angles update:
//   y[i]      = sum_{j set in bits_row[i]} x_in[j]
//   step_out  = (step_in ? step_in[i] : 0) + coef * y[i]
//   x_out[i]  = x_in[i] + step_out[i]
// x_in (16 KB) staged in LDS; double-buffered across hops (no RW race).
// ---------------------------------------------------------------------------
__global__ __launch_bounds__(256)
void hop_mv(const unsigned* __restrict__ bits,
            const float* __restrict__ x_in,
            const float* __restrict__ step_in,
            float coef,
            float* __restrict__ step_out,
            float* __restrict__ x_out)
{
    __shared__ float xs[NNODES];
    const int tid = threadIdx.x;
    for (int i = tid; i < NNODES; i += 256) xs[i] = x_in[i];
    __syncthreads();

    const int wave = tid >> 5;
    const int lane = tid & 31;
    const int row  = blockIdx.x * 8 + wave;

    const uint4 w4 = ((const uint4*)(bits + (size_t)row * NW))[lane];
    unsigned wv[4] = { w4.x, w4.y, w4.z, w4.w };
    float s = 0.f;
    #pragma unroll
    for (int q = 0; q < 4; ++q) {
        unsigned word = wv[q];
        const int base = ((lane * 4 + q) << 5);
        while (word) {
            int b = __builtin_ctz(word);
            word &= word - 1;
            s += xs[base + b];
        }
    }
    for (int o = 16; o; o >>= 1) s += __shfl_down(s, o);
    if (lane == 0) {
        float st = (step_in ? step_in[row] : 0.0f) + coef * s;
        step_out[row] = st;
        x_out[row] = xs[row] + st;
    }
}

// ---------------------------------------------------------------------------
extern "C" void kernel_launch(void* const* d_in, const int* in_sizes, int n_in,
                              void* d_out, int out_size, void* d_ws, size_t ws_size,
                              hipStream_t stream)
{
    (void)in_sizes; (void)n_in; (void)out_size; (void)ws_size;

    const float* coeffs = (const float*)d_in[0];
    const float* W0     = (const float*)d_in[1];
    const float* b0     = (const float*)d_in[2];
    const float* W1     = (const float*)d_in[3];
    const float* W2     = (const float*)d_in[4];
    const float* W3     = (const float*)d_in[5];
    const float* b3     = (const float*)d_in[6];
    const float* gamma  = (const float*)d_in[7];
    const float* beta   = (const float*)d_in[8];
    const int*   ei     = (const int*)d_in[9];
    float* out = (float*)d_out;

    // Workspace layout: 3 x 2MB bitsets + small vectors (~6.4 MB total)
    char* ws = (char*)d_ws;
    const size_t MATB = (size_t)NNODES * NW * sizeof(unsigned);   // 2 MB
    unsigned* A   = (unsigned*)(ws);
    unsigned* A2  = (unsigned*)(ws + MATB);
    unsigned* A3  = (unsigned*)(ws + 2 * MATB);
    float* Xraw   = (float*)(ws + 3 * MATB);
    float* X      = Xraw + NNODES;
    float* sA     = X    + NNODES;
    float* sB     = sA   + NNODES;
    float* angA   = sB   + NNODES;
    float* angB   = angA + NNODES;

    hipMemsetAsync(A, 0, MATB, stream);

    build_adj<<<NEDGES / 256, 256, 0, stream>>>(ei, A);
    mlp_kernel<<<NNODES / 128, 256, 0, stream>>>(coeffs, W0, b0, W1, W2, W3, b3, Xraw);
    bn_kernel<<<1, 1024, 0, stream>>>(Xraw, X, gamma, beta);

    bool_compose<<<NNODES / 8, 256, 0, stream>>>(A, A,  A2);   // A2 = bool(A@A)
    bool_compose<<<NNODES / 8, 256, 0, stream>>>(A, A2, A3);   // A3 = bool(A@A2) == bool(A^3)

    const float L2 = 0.69314718056f;   // log 2
    const float L3 = 1.09861228867f;   // log 3
    const float L4 = 1.38629436112f;   // log 4
    hop_mv<<<NNODES / 8, 256, 0, stream>>>(A,  X,    nullptr, L2, sA, angA);
    hop_mv<<<NNODES / 8, 256, 0, stream>>>(A2, angA, sA,      L3, sB, angB);
    hop_mv<<<NNODES / 8, 256, 0, stream>>>(A3, angB, sB,      L4, sA, out);
}